// CenterLoss_25391846654216
// MI455X (gfx1250) — compile-verified
//
#include <hip/hip_runtime.h>
#include <math.h>

#define CL_EPS  1e-7f
#define CL_BIG  1e11f

typedef __attribute__((ext_vector_type(2))) float v2f;
typedef __attribute__((ext_vector_type(8))) float v8f;

// ---------------------------------------------------------------------------
// Kernel 1: row-wise argmax over predicts [N, C] -> labels[N]
// Bandwidth-dominant pass (~217 MB read). One block per row, coalesced
// strided non-temporal loads (data is single-use; keep it out of the caches).
// Tie-break: first (lowest) index wins, matching jnp.argmax.
// ---------------------------------------------------------------------------
__global__ __launch_bounds__(256) void cl_argmax(const float* __restrict__ predicts,
                                                 int* __restrict__ labels, int C) {
  const int n   = blockIdx.x;
  const int tid = threadIdx.x;
  const float* row = predicts + (size_t)n * (size_t)C;

  float best = -__builtin_inff();
  int   bi   = 0;
  if (tid < C) { best = __builtin_nontemporal_load(row + tid); bi = tid; }
  #pragma unroll 4
  for (int c = tid + 256; c < C; c += 256) {
    float v = __builtin_nontemporal_load(row + c);
    if (v > best) { best = v; bi = c; }   // strict > keeps first occurrence
  }

  __shared__ float sv[256];
  __shared__ int   si[256];
  sv[tid] = best; si[tid] = bi;
  __syncthreads();
  for (int s = 128; s > 0; s >>= 1) {
    if (tid < s) {
      float ov = sv[tid + s]; int oi = si[tid + s];
      float mv = sv[tid];     int mi = si[tid];
      if (ov > mv || (ov == mv && oi < mi)) { sv[tid] = ov; si[tid] = oi; }
    }
    __syncthreads();
  }
  if (tid == 0) labels[n] = si[0];
}

// ---------------------------------------------------------------------------
// Kernel 2: per-wave 16-row tile. A = feats tile (16 x D), B = gathered
// centers (D x 16, column n = centers[label[r0+n]]). Accumulate
// V_WMMA_F32_16X16X4_F32 over K; diagonal of the 16x16 result gives the 16
// dot products. Out-of-range tile rows are CLAMPED to row N-1 (branch-free:
// only the diagonal entries of in-range rows are ever consumed, and diag
// element n depends only on A-row n / B-column n). Squared norms accumulated
// alongside; wave halves combined via shfl_xor(16). Requires D % 4 == 0.
// ---------------------------------------------------------------------------
__global__ __launch_bounds__(256) void cl_dist(const float* __restrict__ feats,
                                               const float* __restrict__ centers,
                                               const int*   __restrict__ labels,
                                               float*       __restrict__ partials,
                                               int N, int D) {
  const int lane = threadIdx.x & 31;
  const int wave = threadIdx.x >> 5;
  const int tile = blockIdx.x * (blockDim.x >> 5) + wave;
  const int r0   = tile * 16;

  const int half = lane >> 4;          // 0: K%4 in {0,1}; 1: K%4 in {2,3}
  const int l16  = lane & 15;          // A-frag row M / B-frag column N
  const int rowA = r0 + l16;
  const int rowC = (rowA < N) ? rowA : (N - 1);   // clamp, no divergence
  const int lbl  = labels[rowC];
  const float* frow = feats   + (size_t)rowC * (size_t)D;
  const float* crow = centers + (size_t)lbl  * (size_t)D;
  const int kofs = half * 2;

  v8f   acc = {};
  float sf = 0.f, sc = 0.f;
  #pragma unroll 4
  for (int kk = 0; kk < D; kk += 4) {
    // 8-byte aligned pair loads (D%4==0 => element offset is even)
    const v2f A  = *(const v2f*)(frow + kk + kofs);
    const v2f Bm = *(const v2f*)(crow + kk + kofs);
    sf = __builtin_fmaf(A.x,  A.x,  sf); sf = __builtin_fmaf(A.y,  A.y,  sf);
    sc = __builtin_fmaf(Bm.x, Bm.x, sc); sc = __builtin_fmaf(Bm.y, Bm.y, sc);
    // 8 args: (neg_a, A, neg_b, B, c_mod, C, reuse_a, reuse_b)
    acc = __builtin_amdgcn_wmma_f32_16x16x4_f32(false, A, false, Bm,
                                                (short)0, acc, false, false);
  }
  // combine the two wave-halves' partial squared norms: lane l then holds the
  // full ||f||^2 and ||c||^2 for tile row (l & 15)
  sf += __shfl_xor(sf, 16, 32);
  sc += __shfl_xor(sc, 16, 32);

  float accArr[8];
  #pragma unroll
  for (int i = 0; i < 8; ++i) accArr[i] = acc[i];

  // diag D[m][m]: m<8 -> VGPR m, lane m ; m>=8 -> VGPR m-8, lane m+16
  float contrib = 0.f;
  const bool active = (lane < 8) || (lane >= 24);
  const int  nrow   = lane & 15;
  const int  row    = r0 + nrow;
  if (active && row < N) {
    float dot = accArr[lane & 7];
    float d   = sf + sc - 2.f * dot;
    d = fminf(fmaxf(d, CL_EPS), CL_BIG);
    contrib = d;
  }

  __shared__ float red[256];
  red[threadIdx.x] = contrib;
  __syncthreads();
  for (int s = 128; s > 0; s >>= 1) {
    if (threadIdx.x < s) red[threadIdx.x] += red[threadIdx.x + s];
    __syncthreads();
  }
  if (threadIdx.x == 0) partials[blockIdx.x] = red[0];
}

// ---------------------------------------------------------------------------
// Kernel 3: deterministic final reduction of per-block partials, write mean.
// ---------------------------------------------------------------------------
__global__ __launch_bounds__(256) void cl_final(const float* __restrict__ partials,
                                                int nPart, float invN,
                                                float* __restrict__ out) {
  __shared__ float red[256];
  float s = 0.f;
  for (int i = threadIdx.x; i < nPart; i += 256) s += partials[i];
  red[threadIdx.x] = s;
  __syncthreads();
  for (int st = 128; st > 0; st >>= 1) {
    if (threadIdx.x < st) red[threadIdx.x] += red[threadIdx.x + st];
    __syncthreads();
  }
  if (threadIdx.x == 0) out[0] = red[0] * invN;
}

extern "C" void kernel_launch(void* const* d_in, const int* in_sizes, int n_in,
                              void* d_out, int out_size, void* d_ws, size_t ws_size,
                              hipStream_t stream) {
  const float* feats    = (const float*)d_in[0];  // [B,T,D] f32
  const float* predicts = (const float*)d_in[1];  // [B,T,C] f32
  const float* centers  = (const float*)d_in[2];  // [C,D]   f32

  // Recover shapes: s0 = N*D, s1 = N*C, s2 = C*D  =>  D = sqrt(s0*s2/s1)
  const long long s0 = in_sizes[0], s1 = in_sizes[1], s2 = in_sizes[2];
  const int D = (int)(sqrt((double)s0 * (double)s2 / (double)s1) + 0.5);
  const int N = (int)(s0 / D);
  const int C = (int)(s2 / D);

  // Workspace: labels[N] ints, then (256B-aligned) per-block partial floats.
  int*   labels = (int*)d_ws;
  size_t pOff   = (((size_t)N * sizeof(int)) + 255) & ~(size_t)255;
  float* partials = (float*)((char*)d_ws + pOff);

  // Pass 1: argmax (bandwidth-dominant, ~9.3 us lower bound at 23.3 TB/s)
  cl_argmax<<<N, 256, 0, stream>>>(predicts, labels, C);

  // Pass 2: WMMA gathered-center distances, 16 rows per wave, 8 waves/block
  const int tiles   = (N + 15) / 16;
  const int blocks2 = (tiles + 7) / 8;
  cl_dist<<<blocks2, 256, 0, stream>>>(feats, centers, labels, partials, N, D);

  // Pass 3: deterministic final reduce + mean
  cl_final<<<1, 256, 0, stream>>>(partials, blocks2, 1.0f / (float)N, (float*)d_out);
}